// GCN_8014408974728
// MI455X (gfx1250) — compile-verified
//
#include <hip/hip_runtime.h>
#include <hip/hip_bf16.h>

typedef __attribute__((ext_vector_type(2))) float v2f;
typedef __attribute__((ext_vector_type(8))) float v8f;

#define BDIM 4
#define NDIM 128
#define IDIM 128
#define DIN  512
#define DOUT 256
#define K1V  20
#define NROWS (BDIM*NDIM*IDIM)   /* 65536 rows of features     */
#define NBN   (BDIM*NDIM)        /* 512 (b,n) pairs            */
#define GROWS (NBN*K1V)          /* 10240 gathered rows        */
#define EPSV  1e-5f

enum { A_BN_CAT = 0, A_CAT = 1, A_PLAIN = 2 };
enum { E_RELU = 0, E_PRELU = 1 };

// ---------------------------------------------------------------------------
// Fused GEMM:  dst[M,NOUT] = act( Atile(src,diag,bn)[M,KTOT] @ W[KTOT,NOUT] + b )
// Atile: for cat modes, columns [0,DINT) are src (optionally BN-normalized),
// columns [DINT,2*DINT) are diag[row] * src (the GraphConv concat trick).
// BM=128, BN=128, KT=16; 8 waves of 32; each wave -> 32x64 via 2x4 WMMA f32
// (64 accumulator VGPRs/lane; frag-load:WMMA ratio 6:8 per k-step group).
// ---------------------------------------------------------------------------
template <int DINT, int KTOT, int NOUT, int AM, int EM>
__global__ __launch_bounds__(256) void gemm_wmma_k(
    const float* __restrict__ src,   // [M, DINT]
    const float* __restrict__ W,     // [KTOT, NOUT]
    const float* __restrict__ bias,  // [NOUT]
    const float* __restrict__ alpha, // [NOUT] (PReLU) or null
    const float* __restrict__ diag,  // [M] or null
    const float* __restrict__ mean,  // [DINT] or null
    const float* __restrict__ invstd,// [DINT] or null
    float* __restrict__ dst)         // [M, NOUT]
{
    __shared__ float ldsA[128 * 18]; // 128 rows x 16 k, stride 18 (8B-aligned v2f reads)
    __shared__ float ldsB[128 * 18]; // 128 cols x 16 k (transposed), stride 18

    const int tid  = threadIdx.x;
    const int lane = tid & 31;
    const int wave = tid >> 5;
    const int wm   = wave & 3;   // 4 waves along M (4*32 = 128)
    const int wn   = wave >> 2;  // 2 waves along N (2*64 = 128)
    const int m0   = blockIdx.y * 128;
    const int n0   = blockIdx.x * 128;
    const int lr   = lane & 15;
    const int hi   = lane >> 4;  // 0: K={0,1} / rows 0-7,  1: K={2,3} / rows 8-15

    v8f acc[2][4] = {};

    for (int k0 = 0; k0 < KTOT; k0 += 16) {
        const bool second = (AM != A_PLAIN) && (k0 >= DINT);
        const int  koff   = second ? (k0 - DINT) : k0;

        // ---- stage A tile: 128x16 = 512 float4 slots, 2 per thread ----
        for (int e = tid; e < 512; e += 256) {
            const int row = e >> 2;
            const int kk4 = (e & 3) << 2;
            const int gr  = m0 + row;
            float4 v = *(const float4*)(src + (size_t)gr * DINT + (koff + kk4));
            if (AM == A_BN_CAT) {
                const float4 mu = *(const float4*)(mean + koff + kk4);
                const float4 is = *(const float4*)(invstd + koff + kk4);
                v.x = (v.x - mu.x) * is.x;
                v.y = (v.y - mu.y) * is.y;
                v.z = (v.z - mu.z) * is.z;
                v.w = (v.w - mu.w) * is.w;
            }
            if (AM != A_PLAIN && second) {
                const float d = diag[gr];
                v.x *= d; v.y *= d; v.z *= d; v.w *= d;
            }
            float* q = &ldsA[row * 18 + kk4];
            q[0] = v.x; q[1] = v.y; q[2] = v.z; q[3] = v.w;
        }
        // ---- stage B tile: 16x128 = 512 float4 slots, stored K-contiguous ----
        for (int e = tid; e < 512; e += 256) {
            const int kk = e >> 5;           // 0..15
            const int c4 = (e & 31) << 2;    // 0..124
            const float4 v = *(const float4*)(W + (size_t)(k0 + kk) * NOUT + n0 + c4);
            ldsB[(c4 + 0) * 18 + kk] = v.x;
            ldsB[(c4 + 1) * 18 + kk] = v.y;
            ldsB[(c4 + 2) * 18 + kk] = v.z;
            ldsB[(c4 + 3) * 18 + kk] = v.w;
        }
        __syncthreads();

        // ---- 4 WMMA k-steps of 4 over this stage ----
#pragma unroll
        for (int ks = 0; ks < 16; ks += 4) {
            v2f af[2], bf[4];
#pragma unroll
            for (int i = 0; i < 2; ++i)
                af[i] = *(const v2f*)&ldsA[(wm * 32 + i * 16 + lr) * 18 + ks + hi * 2];
#pragma unroll
            for (int j = 0; j < 4; ++j)
                bf[j] = *(const v2f*)&ldsB[(wn * 64 + j * 16 + lr) * 18 + ks + hi * 2];
#pragma unroll
            for (int i = 0; i < 2; ++i)
#pragma unroll
                for (int j = 0; j < 4; ++j)
                    acc[i][j] = __builtin_amdgcn_wmma_f32_16x16x4_f32(
                        false, af[i], false, bf[j], (short)0, acc[i][j], false, false);
        }
        __syncthreads();
    }

    // ---- epilogue: bias + ReLU / PReLU, write out ----
#pragma unroll
    for (int i = 0; i < 2; ++i) {
#pragma unroll
        for (int j = 0; j < 4; ++j) {
            const int   col  = n0 + wn * 64 + j * 16 + lr;
            const float bcol = bias[col];
            const float acol = (EM == E_PRELU) ? alpha[col] : 0.0f;
#pragma unroll
            for (int v = 0; v < 8; ++v) {
                const int row = m0 + wm * 32 + i * 16 + hi * 8 + v;
                float x = acc[i][j][v] + bcol;
                if (EM == E_RELU) x = fmaxf(x, 0.0f);
                else              x = (x >= 0.0f) ? x : acol * x;
                dst[(size_t)row * NOUT + col] = x;
            }
        }
    }
}

// ---------------------------------------------------------------------------
// Helpers
// ---------------------------------------------------------------------------
__global__ void zero_f32(float* p, int n) {
    int i = blockIdx.x * blockDim.x + threadIdx.x;
    if (i < n) p[i] = 0.0f;
}

__global__ __launch_bounds__(256) void bn_partial(const float* __restrict__ x,
                                                  float* __restrict__ sums,
                                                  float* __restrict__ ssqs) {
    const int col = blockIdx.x * 256 + threadIdx.x;   // 0..511
    const int r0  = blockIdx.y * 1024;
    float s = 0.0f, ss = 0.0f;
    for (int r = r0; r < r0 + 1024; ++r) {
        const float v = x[(size_t)r * DIN + col];
        s += v; ss += v * v;
    }
    atomicAdd(&sums[col], s);
    atomicAdd(&ssqs[col], ss);
}

__global__ void bn_finalize(const float* __restrict__ sums, const float* __restrict__ ssqs,
                            float* __restrict__ mean, float* __restrict__ invstd) {
    const int c = blockIdx.x * blockDim.x + threadIdx.x;
    if (c < DIN) {
        const float m   = sums[c] * (1.0f / NROWS);
        const float var = ssqs[c] * (1.0f / NROWS) - m * m;  // biased, like torch BN
        mean[c]   = m;
        invstd[c] = rsqrtf(var + EPSV);
    }
}

__global__ void diag_extract(const float* __restrict__ A, float* __restrict__ diag) {
    const int r = blockIdx.x * blockDim.x + threadIdx.x;  // r = bn*I + i
    if (r < NROWS) {
        const int bn = r >> 7;
        const int i  = r & 127;
        diag[r] = A[(size_t)bn * IDIM * IDIM + (size_t)i * IDIM + i];
    }
}

__global__ void build_idx(const unsigned char* __restrict__ mask, int* __restrict__ idx) {
    const int bn = blockIdx.x * blockDim.x + threadIdx.x;
    if (bn < NBN) {
        int cnt = 0;
        for (int i = 0; i < IDIM && cnt < K1V; ++i)
            if (mask[bn * IDIM + i]) { idx[bn * K1V + cnt] = i; ++cnt; }
        for (; cnt < K1V; ++cnt) idx[bn * K1V + cnt] = 0; // defensive
    }
}

__global__ __launch_bounds__(256) void gather_rows(const float* __restrict__ h,
                                                   const int* __restrict__ idx,
                                                   float* __restrict__ g) {
    const int grow = blockIdx.x;        // 0..GROWS-1
    const int c    = threadIdx.x;       // 0..255
    const int bn   = grow / K1V;
    const int k    = grow - bn * K1V;
    const int srow = bn * IDIM + idx[bn * K1V + k];
    g[(size_t)grow * DOUT + c] = h[(size_t)srow * DOUT + c];
}

__global__ __launch_bounds__(256) void final_linear(const float* __restrict__ z,
                                                    const float* __restrict__ Wc2,
                                                    const float* __restrict__ bc2,
                                                    float* __restrict__ out) {
    const int g = blockIdx.x * blockDim.x + threadIdx.x;
    if (g < GROWS) {
        float a0 = bc2[0], a1 = bc2[1];
        const float* zr = z + (size_t)g * DOUT;
        for (int d = 0; d < DOUT; ++d) {
            const float v = zr[d];
            a0 += v * Wc2[d * 2 + 0];
            a1 += v * Wc2[d * 2 + 1];
        }
        out[g * 2 + 0] = a0;
        out[g * 2 + 1] = a1;
    }
}

// ---------------------------------------------------------------------------
// Launch
// ---------------------------------------------------------------------------
extern "C" void kernel_launch(void* const* d_in, const int* in_sizes, int n_in,
                              void* d_out, int out_size, void* d_ws, size_t ws_size,
                              hipStream_t stream) {
    const float* x    = (const float*)d_in[0];
    const float* A    = (const float*)d_in[1];
    const unsigned char* mask = (const unsigned char*)d_in[2];
    /* d_in[3] = k1 scalar (K1V hardcoded to match reference) */
    const float* W1 = (const float*)d_in[4];  const float* b1 = (const float*)d_in[5];
    const float* W2 = (const float*)d_in[6];  const float* b2 = (const float*)d_in[7];
    const float* W3 = (const float*)d_in[8];  const float* b3 = (const float*)d_in[9];
    const float* W4 = (const float*)d_in[10]; const float* b4 = (const float*)d_in[11];
    const float* Wc1 = (const float*)d_in[12]; const float* bc1 = (const float*)d_in[13];
    const float* alpha = (const float*)d_in[14];
    const float* Wc2 = (const float*)d_in[15]; const float* bc2 = (const float*)d_in[16];
    float* out = (float*)d_out;

    float* ws     = (float*)d_ws;
    float* sums   = ws;                       // 512
    float* ssqs   = ws + 512;                 // 512
    float* mean   = ws + 1024;                // 512
    float* invstd = ws + 1536;                // 512
    float* diag   = ws + 2048;                // 65536
    int*   idx    = (int*)(ws + 2048 + NROWS);// 10240 ints
    float* bufA   = ws + (1 << 17);           // 65536*512 f32
    float* bufB   = bufA + (size_t)NROWS * DIN;
    float* gbuf   = bufA;                     // overlay: h3 (in bufA) dead by then
    float* zbuf   = bufA + (size_t)GROWS * DOUT;

    zero_f32<<<dim3(4), dim3(256), 0, stream>>>(sums, 1024);           // sums+ssqs
    bn_partial<<<dim3(2, 64), dim3(256), 0, stream>>>(x, sums, ssqs);
    bn_finalize<<<dim3(2), dim3(256), 0, stream>>>(sums, ssqs, mean, invstd);
    diag_extract<<<dim3(NROWS / 256), dim3(256), 0, stream>>>(A, diag);

    // Layer 1: BN(x) concat-GEMM, K=1024, N=512
    gemm_wmma_k<DIN, 2 * DIN, DIN, A_BN_CAT, E_RELU>
        <<<dim3(DIN / 128, NROWS / 128), 256, 0, stream>>>(x, W1, b1, nullptr, diag, mean, invstd, bufA);
    // Layer 2: K=1024, N=512
    gemm_wmma_k<DIN, 2 * DIN, DIN, A_CAT, E_RELU>
        <<<dim3(DIN / 128, NROWS / 128), 256, 0, stream>>>(bufA, W2, b2, nullptr, diag, nullptr, nullptr, bufB);
    // Layer 3: K=1024, N=256
    gemm_wmma_k<DIN, 2 * DIN, DOUT, A_CAT, E_RELU>
        <<<dim3(DOUT / 128, NROWS / 128), 256, 0, stream>>>(bufB, W3, b3, nullptr, diag, nullptr, nullptr, bufA);
    // Layer 4: K=512, N=256
    gemm_wmma_k<DOUT, 2 * DOUT, DOUT, A_CAT, E_RELU>
        <<<dim3(DOUT / 128, NROWS / 128), 256, 0, stream>>>(bufA, W4, b4, nullptr, diag, nullptr, nullptr, bufB);

    build_idx<<<dim3(2), dim3(256), 0, stream>>>(mask, idx);
    gather_rows<<<dim3(GROWS), dim3(256), 0, stream>>>(bufB, idx, gbuf);

    // Classifier layer 1: 10240 x 256 x 256, PReLU epilogue
    gemm_wmma_k<DOUT, DOUT, DOUT, A_PLAIN, E_PRELU>
        <<<dim3(DOUT / 128, GROWS / 128), 256, 0, stream>>>(gbuf, Wc1, bc1, alpha, nullptr, nullptr, nullptr, zbuf);

    final_linear<<<dim3(GROWS / 256), dim3(256), 0, stream>>>(zbuf, Wc2, bc2, out);

    (void)in_sizes; (void)n_in; (void)out_size; (void)ws_size;
}